// Proposed_ver1_21071109554385
// MI455X (gfx1250) — compile-verified
//
#include <hip/hip_runtime.h>

// Problem constants fixed by the reference setup_inputs():
#define N_   64
#define C_   256
#define H_   56
#define W_   56
#define G_   32
#define HW_  (H_ * W_)          // 3136
#define K2N_ (2 * N_)           // 128  (features per channel: N means + N vars)
#define EPS_ 1e-5f

typedef __attribute__((ext_vector_type(2))) float v2f;
typedef __attribute__((ext_vector_type(4))) float v4f;
typedef __attribute__((ext_vector_type(8))) float v8f;

// ---------------------------------------------------------------------------
// Kernel 1: per-(n,c) plane reduction: s1 = sum(x), s2 = sum(x*x)
// One block per plane (N*C = 16384 blocks). float4 (global_load_b128) loads.
// Temporal loads on purpose: x (~205MB) nearly fits the 192MB L2, warming it
// for the apply pass.
// ---------------------------------------------------------------------------
__global__ __launch_bounds__(256) void k_stats(const float* __restrict__ x,
                                               float* __restrict__ s1,
                                               float* __restrict__ s2) {
    const int plane = blockIdx.x;                       // n*C + c
    const v4f* xp = (const v4f*)(x + (size_t)plane * HW_);
    const int nvec = HW_ / 4;                           // 784
    float sum = 0.f, sq = 0.f;
    for (int i = threadIdx.x; i < nvec; i += 256) {
        v4f v = xp[i];
        sum += (v.x + v.y) + (v.z + v.w);
        sq  += (v.x * v.x + v.y * v.y) + (v.z * v.z + v.w * v.w);
    }
    __shared__ float red1[256];
    __shared__ float red2[256];
    red1[threadIdx.x] = sum;
    red2[threadIdx.x] = sq;
    __syncthreads();
    for (int off = 128; off > 0; off >>= 1) {
        if (threadIdx.x < off) {
            red1[threadIdx.x] += red1[threadIdx.x + off];
            red2[threadIdx.x] += red2[threadIdx.x + off];
        }
        __syncthreads();
    }
    if (threadIdx.x == 0) {
        s1[plane] = red1[0];
        s2[plane] = red2[0];
    }
}

// s[c][j]: j < N -> mean_cn[c][j]; j >= N -> var_cn[c][j-N] (ddof=1)
__device__ __forceinline__ float sval(int c, int j,
                                      const float* __restrict__ s1g,
                                      const float* __restrict__ s2g,
                                      float inv_hw, float inv_hwm1) {
    if (j < N_) {
        return s1g[j * C_ + c] * inv_hw;
    } else {
        const int n = j - N_;
        const float S1v = s1g[n * C_ + c];
        const float m1  = S1v * inv_hw;
        return (s2g[n * C_ + c] - S1v * m1) * inv_hwm1;
    }
}

// ---------------------------------------------------------------------------
// Kernel 2 (single block, 512 threads = 16 waves):
//   logits(256x32) = s(256x128) @ fc_w^T + fc_b   via V_WMMA_F32_16X16X4_F32
//   g[c] = argmax_g logits[c][g]   (softmax is monotone -> same argmax)
//   group aggregation -> mean_g/var_g (ddof-style denominators per reference)
//   emit per-(n,c) affine:  a = rsqrt(var+eps)*weight[c],  b = bias[c] - mean*a
// ---------------------------------------------------------------------------
__global__ __launch_bounds__(512) void k_group(const float* __restrict__ s1g,
                                               const float* __restrict__ s2g,
                                               const float* __restrict__ fcw,   // (G, 2N)
                                               const float* __restrict__ fcb,   // (G,)
                                               const float* __restrict__ weight,// (C,)
                                               const float* __restrict__ bias,  // (C,)
                                               float* __restrict__ aout,        // (N*C,)
                                               float* __restrict__ bout) {      // (N*C,)
    __shared__ float logits[C_][G_];
    __shared__ int   gidx[C_];
    __shared__ float S1s[N_ * G_];
    __shared__ float S2s[N_ * G_];
    __shared__ float cnts[G_];

    const int tid  = threadIdx.x;
    const int lane = tid & 31;
    const int wave = tid >> 5;           // 0..15 : row-tile of C
    const float inv_hw   = 1.0f / (float)HW_;
    const float inv_hwm1 = 1.0f / (float)(HW_ - 1);

    // ---- WMMA matmul: each wave computes rows [wave*16, wave*16+16) x all 32 cols
    // A 16x4 f32 layout: lanes 0-15 hold (K=k0,k0+1), lanes 16-31 hold (K=k0+2,k0+3)
    // B 4x16 f32 layout: transposed analog (row K striped across lanes within a VGPR)
    const int m     = lane & 15;
    const int row   = wave * 16 + m;
    const int khalf = (lane >> 4) << 1;  // 0 for lanes 0-15, 2 for lanes 16-31

    v8f acc0 = {};  // columns 0..15
    v8f acc1 = {};  // columns 16..31
    for (int k0 = 0; k0 < K2N_; k0 += 4) {
        const int k = k0 + khalf;
        v2f afrag, b0, b1;
        afrag.x = sval(row, k,     s1g, s2g, inv_hw, inv_hwm1);
        afrag.y = sval(row, k + 1, s1g, s2g, inv_hw, inv_hwm1);
        b0.x = fcw[m * K2N_ + k];
        b0.y = fcw[m * K2N_ + k + 1];
        b1.x = fcw[(m + 16) * K2N_ + k];
        b1.y = fcw[(m + 16) * K2N_ + k + 1];
        acc0 = __builtin_amdgcn_wmma_f32_16x16x4_f32(false, afrag, false, b0,
                                                     (short)0, acc0, false, false);
        acc1 = __builtin_amdgcn_wmma_f32_16x16x4_f32(false, afrag, false, b1,
                                                     (short)0, acc1, false, false);
    }
    // D layout: VGPR r -> M = r (lanes 0-15) / r+8 (lanes 16-31), N = lane&15
    {
        const int rofs = (lane >> 4) * 8;
        const float fb0 = fcb[m];
        const float fb1 = fcb[m + 16];
#pragma unroll
        for (int r = 0; r < 8; ++r) {
            logits[wave * 16 + r + rofs][m]      = acc0[r] + fb0;
            logits[wave * 16 + r + rofs][m + 16] = acc1[r] + fb1;
        }
    }
    __syncthreads();

    // ---- per-channel argmax (first max wins, matching jnp.argmax)
    if (tid < C_) {
        float best = logits[tid][0];
        int bg = 0;
#pragma unroll
        for (int g = 1; g < G_; ++g) {
            const float l = logits[tid][g];
            if (l > best) { best = l; bg = g; }
        }
        gidx[tid] = bg;
    }
    for (int i = tid; i < N_ * G_; i += 512) { S1s[i] = 0.f; S2s[i] = 0.f; }
    if (tid < G_) cnts[tid] = 0.f;
    __syncthreads();

    // ---- group aggregation
    if (tid < C_) atomicAdd(&cnts[gidx[tid]], 1.0f);
    for (int i = tid; i < N_ * C_; i += 512) {
        const int n = i >> 8;            // C_ == 256
        const int c = i & (C_ - 1);
        const int g = gidx[c];
        atomicAdd(&S1s[n * G_ + g], s1g[i]);
        atomicAdd(&S2s[n * G_ + g], s2g[i]);
    }
    __syncthreads();

    // ---- group mean/var (reference: cnt = nch*hw; var = (S2 - S1*mean)/max(cnt-1,1))
    for (int i = tid; i < N_ * G_; i += 512) {
        const float cf  = cnts[i & (G_ - 1)] * (float)HW_;
        const float S1v = S1s[i];
        const float S2v = S2s[i];
        const float mg  = S1v / fmaxf(cf, 1.0f);
        const float vg  = (S2v - S1v * mg) / fmaxf(cf - 1.0f, 1.0f);
        S1s[i] = mg;
        S2s[i] = vg;
    }
    __syncthreads();

    // ---- per-(n,c) affine coefficients
    for (int i = tid; i < N_ * C_; i += 512) {
        const int n = i >> 8;
        const int c = i & (C_ - 1);
        const int g = gidx[c];
        const float mg = S1s[n * G_ + g];
        const float vg = S2s[n * G_ + g];
        const float av = rsqrtf(vg + EPS_) * weight[c];
        const float bv = bias[c] - mg * av;
        aout[i] = av;
        bout[i] = bv;
    }
}

// ---------------------------------------------------------------------------
// Kernel 3: out = x * a[plane] + b[plane].  float4 streams; NT load of x
// (last use) and NT store of out so the streams don't thrash L2.
// ---------------------------------------------------------------------------
__global__ __launch_bounds__(256) void k_apply(const float* __restrict__ x,
                                               const float* __restrict__ a,
                                               const float* __restrict__ b,
                                               float* __restrict__ out) {
    const int i = blockIdx.x * 256 + threadIdx.x;      // vec4 index; grid sized exactly
    const int plane = i / (HW_ / 4);                   // n*C + c
    const float av = a[plane];
    const float bv = b[plane];
    v4f v = __builtin_nontemporal_load((const v4f*)x + i);
    v4f o;
    o.x = v.x * av + bv;
    o.y = v.y * av + bv;
    o.z = v.z * av + bv;
    o.w = v.w * av + bv;
    __builtin_nontemporal_store(o, (v4f*)out + i);
}

extern "C" void kernel_launch(void* const* d_in, const int* in_sizes, int n_in,
                              void* d_out, int out_size, void* d_ws, size_t ws_size,
                              hipStream_t stream) {
    const float* x      = (const float*)d_in[0];
    const float* fcw    = (const float*)d_in[1];
    const float* fcb    = (const float*)d_in[2];
    const float* weight = (const float*)d_in[3];
    const float* bias   = (const float*)d_in[4];
    // d_in[5] = group (32), fixed at compile time.

    float* s1 = (float*)d_ws;                 // N*C
    float* s2 = s1 + N_ * C_;                 // N*C
    float* aw = s2 + N_ * C_;                 // N*C
    float* bw = aw + N_ * C_;                 // N*C

    k_stats<<<N_ * C_, 256, 0, stream>>>(x, s1, s2);
    k_group<<<1, 512, 0, stream>>>(s1, s2, fcw, fcb, weight, bias, aw, bw);

    const int nvec = N_ * C_ * (HW_ / 4);     // 12,845,056 -> 50,176 blocks exactly
    k_apply<<<nvec / 256, 256, 0, stream>>>(x, aw, bw, (float*)d_out);
}